// NeighborSearch_77721728188839
// MI455X (gfx1250) — compile-verified
//
#include <hip/hip_runtime.h>

typedef __attribute__((ext_vector_type(2))) float v2f;
typedef __attribute__((ext_vector_type(8))) float v8f;

static constexpr int QT = 16;      // query rows per tile (WMMA M)
static constexpr int NCHUNK = 16;  // data-dimension chunks for parallelism
#define EPSQ (1e-7f * 1e-7f)

// ---------------------------------------------------------------------------
// Pass 1: count neighbors per (query row, data chunk) using WMMA f32 16x16x4.
// D = A x B + C with A row = [-2qx,-2qy,-2qz,1], B col = [dx,dy,dz,||d||^2],
// C[m][*] = ||q_m||^2  ==>  D[m][n] = squared distance.
// ---------------------------------------------------------------------------
__global__ __launch_bounds__(32)
void nbr_count(const float* __restrict__ data, const float* __restrict__ qry,
               const float* __restrict__ radius, int N, int M,
               int* __restrict__ cnts)
{
    const int m0    = blockIdx.x * QT;
    const int chunk = blockIdx.y;
    const int lane  = threadIdx.x;
    const int half  = lane >> 4;   // 0: K slots 0,1 / rows i ; 1: K slots 2,3 / rows i+8
    const int nl    = lane & 15;

    const float r  = radius[0];
    const float r2 = r * r;

    // A fragment (16x4 f32): lanes 0-15 hold K0,K1 of row nl; lanes 16-31 hold K2,K3
    v2f a;
    {
        const float* qp = qry + (size_t)(m0 + nl) * 3;
        if (half == 0) { a.x = -2.0f * qp[0]; a.y = -2.0f * qp[1]; }
        else           { a.x = -2.0f * qp[2]; a.y = 1.0f; }
    }
    // C fragment: VGPR i -> row m0 + i + 8*half; broadcast ||q||^2 across columns
    v8f c;
#pragma unroll
    for (int i = 0; i < 8; ++i) {
        const float* qp = qry + (size_t)(m0 + i + 8 * half) * 3;
        c[i] = qp[0] * qp[0] + qp[1] * qp[1] + qp[2] * qp[2];
    }

    int cntlo[8], cnthi[8];
#pragma unroll
    for (int i = 0; i < 8; ++i) { cntlo[i] = 0; cnthi[i] = 0; }

    const int span = N / NCHUNK;
    const int jbeg = chunk * span;
    const int jend = jbeg + span;
    for (int j0 = jbeg; j0 < jend; j0 += 16) {
        const float* dp = data + (size_t)(j0 + nl) * 3;
        __builtin_prefetch(dp + 16 * 3, 0, 1);      // next tile -> global_prefetch_b8
        float dx = dp[0], dy = dp[1], dz = dp[2];
        // B fragment (4x16 f32): lanes 0-15 K0,K1 of col nl; lanes 16-31 K2,K3
        v2f b;
        if (half == 0) { b.x = dx; b.y = dy; }
        else           { b.x = dz; b.y = dx * dx + dy * dy + dz * dz; }

        v8f sq = __builtin_amdgcn_wmma_f32_16x16x4_f32(false, a, false, b,
                                                       (short)0, c, false, false);
#pragma unroll
        for (int i = 0; i < 8; ++i) {
            unsigned msk = __builtin_amdgcn_ballot_w32(sq[i] <= r2);
            cntlo[i] += __builtin_popcount(msk & 0xFFFFu);  // row i, cols 0-15
            cnthi[i] += __builtin_popcount(msk >> 16);      // row i+8
        }
    }

    if (lane == 0) {
#pragma unroll
        for (int i = 0; i < 8; ++i) {
            cnts[(size_t)(m0 + i) * NCHUNK + chunk]     = cntlo[i];
            cnts[(size_t)(m0 + i + 8) * NCHUNK + chunk] = cnthi[i];
        }
    }
}

// ---------------------------------------------------------------------------
// Pass 2: in-place exclusive scan of n counts; arr[n] = grand total K.
// ---------------------------------------------------------------------------
__global__ __launch_bounds__(256)
void nbr_scan(int* __restrict__ arr, int n)
{
    __shared__ int part[256];
    const int t  = threadIdx.x;
    const int ch = (n + 255) / 256;
    const int b  = t * ch;
    const int e  = (b + ch < n) ? (b + ch) : n;

    int s = 0;
    for (int i = b; i < e; ++i) s += arr[i];
    part[t] = s;
    __syncthreads();
    for (int off = 1; off < 256; off <<= 1) {
        int v = (t >= off) ? part[t - off] : 0;
        __syncthreads();
        if (t >= off) part[t] += v;
        __syncthreads();
    }
    int run = (t == 0) ? 0 : part[t - 1];
    for (int i = b; i < e; ++i) { int v = arr[i]; arr[i] = run; run += v; }
    if (t == 255) arr[n] = run;
}

// ---------------------------------------------------------------------------
// Pass 3: write row_splits into d_out at element offset K (device-known).
// ---------------------------------------------------------------------------
__global__ __launch_bounds__(256)
void nbr_splits(const int* __restrict__ scan, int M, float* __restrict__ out)
{
    int i = blockIdx.x * blockDim.x + threadIdx.x;
    int K = scan[(size_t)M * NCHUNK];
    int* osp = (int*)out + K;
    if (i < M)       osp[i] = scan[(size_t)i * NCHUNK];
    else if (i == M) osp[M] = K;
}

// ---------------------------------------------------------------------------
// Pass 4: recompute distances with WMMA, compact indices + weights.
// ---------------------------------------------------------------------------
__global__ __launch_bounds__(32)
void nbr_fill(const float* __restrict__ data, const float* __restrict__ qry,
              const float* __restrict__ radius, int N, int M,
              const int* __restrict__ scan, float* __restrict__ out)
{
    const int m0    = blockIdx.x * QT;
    const int chunk = blockIdx.y;
    const int lane  = threadIdx.x;
    const int half  = lane >> 4;
    const int nl    = lane & 15;

    const float r  = radius[0];
    const float r2 = r * r;
    const int   K  = scan[(size_t)M * NCHUNK];

    int*   out_idx = (int*)out;                       // [0, K)
    float* out_w   = out + (size_t)K + M + 1;         // after splits region

    v2f a;
    {
        const float* qp = qry + (size_t)(m0 + nl) * 3;
        if (half == 0) { a.x = -2.0f * qp[0]; a.y = -2.0f * qp[1]; }
        else           { a.x = -2.0f * qp[2]; a.y = 1.0f; }
    }
    v8f c;
#pragma unroll
    for (int i = 0; i < 8; ++i) {
        const float* qp = qry + (size_t)(m0 + i + 8 * half) * 3;
        c[i] = qp[0] * qp[0] + qp[1] * qp[1] + qp[2] * qp[2];
    }

    int baselo[8], basehi[8];
#pragma unroll
    for (int i = 0; i < 8; ++i) {
        baselo[i] = scan[(size_t)(m0 + i) * NCHUNK + chunk];
        basehi[i] = scan[(size_t)(m0 + i + 8) * NCHUNK + chunk];
    }

    const int span = N / NCHUNK;
    const int jbeg = chunk * span;
    const int jend = jbeg + span;
    const unsigned below = (1u << nl) - 1u;   // lanes below me within my half
    for (int j0 = jbeg; j0 < jend; j0 += 16) {
        const float* dp = data + (size_t)(j0 + nl) * 3;
        __builtin_prefetch(dp + 16 * 3, 0, 1);
        float dx = dp[0], dy = dp[1], dz = dp[2];
        v2f b;
        if (half == 0) { b.x = dx; b.y = dy; }
        else           { b.x = dz; b.y = dx * dx + dy * dy + dz * dz; }

        v8f sq = __builtin_amdgcn_wmma_f32_16x16x4_f32(false, a, false, b,
                                                       (short)0, c, false, false);
#pragma unroll
        for (int i = 0; i < 8; ++i) {
            float    sqv  = sq[i];
            bool     pred = sqv <= r2;
            unsigned msk  = __builtin_amdgcn_ballot_w32(pred);
            unsigned mlo  = msk & 0xFFFFu;
            unsigned mhi  = msk >> 16;
            unsigned mh   = half ? mhi : mlo;
            int      base = half ? basehi[i] : baselo[i];
            if (pred) {
                int pos = base + __builtin_popcount(mh & below);
                out_idx[pos] = j0 + nl;
                float sc = fmaxf(sqv, 0.0f);
                out_w[pos] = (sc == 0.0f) ? EPSQ : sc;   // dist==0 -> EPS -> EPS^2
            }
            baselo[i] += __builtin_popcount(mlo);
            basehi[i] += __builtin_popcount(mhi);
        }
    }
}

// ---------------------------------------------------------------------------
extern "C" void kernel_launch(void* const* d_in, const int* in_sizes, int n_in,
                              void* d_out, int out_size, void* d_ws, size_t ws_size,
                              hipStream_t stream)
{
    (void)n_in; (void)out_size; (void)ws_size;
    const float* data = (const float*)d_in[0];
    const float* qry  = (const float*)d_in[1];
    const float* rad  = (const float*)d_in[2];
    const int N = in_sizes[0] / 3;   // 32768
    const int M = in_sizes[1] / 3;   // 8192

    int* cnts = (int*)d_ws;          // M*NCHUNK + 1 ints, scanned in place

    dim3 grid(M / QT, NCHUNK);
    nbr_count<<<grid, 32, 0, stream>>>(data, qry, rad, N, M, cnts);
    nbr_scan<<<1, 256, 0, stream>>>(cnts, M * NCHUNK);
    nbr_splits<<<((M + 1) + 255) / 256, 256, 0, stream>>>(cnts, M, (float*)d_out);
    nbr_fill<<<grid, 32, 0, stream>>>(data, qry, rad, N, M, cnts, (float*)d_out);
}